// ButterflyRotation_46789373722664
// MI455X (gfx1250) — compile-verified
//
#include <hip/hip_runtime.h>
#include <stdint.h>
#include <math.h>

#define DIMX 4096
#define NLAYERS 12
#define HALF 2048
#define NANG (NLAYERS * HALF)     // 24576 angles
#define TPB 256
#define ROWS_PER_WG 8
#define ROW_STRIDE 4352           // 4096 + 16 pad floats per 256-float segment

// padded LDS index for row element d: +16 floats per 256-float segment
__device__ __forceinline__ unsigned pad_idx(unsigned d) {
    return d + ((d >> 8) << 4);
}

// Base (non-phase) bits of the element index owned by thread t in each phase.
// Phase PB owns bits [PB, PB+3] via the register index e.
template <int PB>
__device__ __forceinline__ unsigned dbase_of(unsigned t) {
    if (PB == 0) return t << 4;                          // e in bits 0..3
    if (PB == 4) return ((t >> 4) << 8) | (t & 15u);     // e in bits 4..7
    return t;                                            // PB==8, e in bits 8..11
}

// q-th register index with bit m clear (pair "low" element)
__device__ __forceinline__ constexpr int pair_lo(int m, int q) {
    return ((q >> m) << (m + 1)) | (q & ((1 << m) - 1));
}

// ---------------- cos/sin table precompute ----------------
__global__ void __launch_bounds__(TPB)
cs_table_kernel(const float* __restrict__ ang, float* __restrict__ ct,
                float* __restrict__ st, int n) {
    int i = blockIdx.x * blockDim.x + threadIdx.x;
    if (i < n) {
        float sv, cv;
        sincosf(ang[i], &sv, &cv);
        ct[i] = cv;
        st[i] = sv;
    }
}

// Load the 32 (c,s) pairs this thread needs for one 4-layer phase.
template <int PB, bool USET>
__device__ __forceinline__ void load_cs(unsigned t,
                                        const float* __restrict__ ct,
                                        const float* __restrict__ st,
                                        const float* __restrict__ ang,
                                        float (&c)[4][8], float (&s)[4][8]) {
    const unsigned db = dbase_of<PB>(t);
#pragma unroll
    for (int m = 0; m < 4; ++m) {
        const int l = PB + m;
#pragma unroll
        for (int q = 0; q < 8; ++q) {
            const int e0 = pair_lo(m, q);
            unsigned d = db | ((unsigned)e0 << PB);
            // angle pair index = d with bit l deleted
            unsigned p = ((d >> (l + 1)) << l) | (d & ((1u << l) - 1u));
            unsigned idx = (unsigned)l * HALF + p;
            if (USET) {
                c[m][q] = ct[idx];
                s[m][q] = st[idx];
            } else {
                float sv, cv;
                sincosf(ang[idx], &sv, &cv);
                c[m][q] = cv;
                s[m][q] = sv;
            }
        }
    }
}

// 4 butterfly layers over 16 register-resident elements (bits 0..3 of e).
__device__ __forceinline__ void rotate16(float (&xv)[16],
                                         const float (&c)[4][8],
                                         const float (&s)[4][8]) {
#pragma unroll
    for (int m = 0; m < 4; ++m) {
#pragma unroll
        for (int q = 0; q < 8; ++q) {
            const int e0 = pair_lo(m, q);
            const int e1 = e0 | (1 << m);
            float xi = xv[e0], xj = xv[e1];
            xv[e0] = fmaf(c[m][q], xi, s[m][q] * xj);
            xv[e1] = fmaf(c[m][q], xj, -(s[m][q] * xi));
        }
    }
}

template <int PB>
__device__ __forceinline__ void lds_read16(const float* row, unsigned t,
                                           float (&xv)[16]) {
    if (PB == 0) {
        unsigned i0 = pad_idx(t << 4);   // contiguous 16 floats
        const float4* p = (const float4*)(row + i0);
#pragma unroll
        for (int j = 0; j < 4; ++j) {
            float4 v = p[j];
            xv[4 * j + 0] = v.x; xv[4 * j + 1] = v.y;
            xv[4 * j + 2] = v.z; xv[4 * j + 3] = v.w;
        }
    } else {
        const unsigned db = dbase_of<PB>(t);
#pragma unroll
        for (int e = 0; e < 16; ++e)
            xv[e] = row[pad_idx(db | ((unsigned)e << PB))];
    }
}

template <int PB>
__device__ __forceinline__ void lds_write16(float* row, unsigned t,
                                            const float (&xv)[16]) {
    if (PB == 0) {
        unsigned i0 = pad_idx(t << 4);
        float4* p = (float4*)(row + i0);
#pragma unroll
        for (int j = 0; j < 4; ++j) {
            float4 v;
            v.x = xv[4 * j + 0]; v.y = xv[4 * j + 1];
            v.z = xv[4 * j + 2]; v.w = xv[4 * j + 3];
            p[j] = v;
        }
    } else {
        const unsigned db = dbase_of<PB>(t);
#pragma unroll
        for (int e = 0; e < 16; ++e)
            row[pad_idx(db | ((unsigned)e << PB))] = xv[e];
    }
}

// ---------------- main fused butterfly kernel ----------------
template <bool USET>
__global__ void __launch_bounds__(TPB)
butterfly_kernel(const float* __restrict__ x, float* __restrict__ out,
                 const float* __restrict__ ct, const float* __restrict__ st,
                 const float* __restrict__ ang, int nrows) {
    extern __shared__ float lds[];
    const unsigned t = threadIdx.x;
    const unsigned row0 = blockIdx.x * ROWS_PER_WG;
    const long long gbase = (long long)row0 * DIMX;
    const float* src = x + gbase;

    int rmax = nrows - (int)row0;
    if (rmax > ROWS_PER_WG) rmax = ROWS_PER_WG;

    // low 32 bits of a generic pointer to LDS == workgroup LDS byte offset
    const unsigned ldsBase = (unsigned)(uintptr_t)(&lds[0]);

    // -------- async global -> LDS stream-in (CDNA5 path, ASYNCcnt) --------
    // x is streamed once: non-temporal so it doesn't evict the c/s tables.
#pragma unroll
    for (int i = 0; i < (ROWS_PER_WG * (DIMX / 4)) / TPB; ++i) {  // 32 iters
        unsigned k = t + TPB * (unsigned)i;       // float4 chunk id
        unsigned r = k >> 10;                     // / (DIMX/4)
        if ((int)r < rmax) {
            unsigned d = (k & 1023u) << 2;        // element within row
            unsigned li = r * ROW_STRIDE + pad_idx(d);
            unsigned lAddr = ldsBase + li * 4u;   // LDS byte address
            unsigned gOff = k << 4;               // global byte offset
            asm volatile("global_load_async_to_lds_b128 %0, %1, %2 th:TH_LOAD_NT"
                         :: "v"(lAddr), "v"(gOff), "s"(src) : "memory");
        }
    }
    asm volatile("s_wait_asynccnt 0x0" ::: "memory");
    __syncthreads();

    float c[4][8], s[4][8], xv[16];

    // -------- phase A: layers 0..3 (element bits 0..3 in registers) --------
    load_cs<0, USET>(t, ct, st, ang, c, s);
    for (int r = 0; r < rmax; ++r) {
        float* rp = lds + (unsigned)r * ROW_STRIDE;
        lds_read16<0>(rp, t, xv);
        rotate16(xv, c, s);
        lds_write16<0>(rp, t, xv);
    }
    __syncthreads();

    // -------- phase B: layers 4..7 --------
    load_cs<4, USET>(t, ct, st, ang, c, s);
    for (int r = 0; r < rmax; ++r) {
        float* rp = lds + (unsigned)r * ROW_STRIDE;
        lds_read16<4>(rp, t, xv);
        rotate16(xv, c, s);
        lds_write16<4>(rp, t, xv);
    }
    __syncthreads();

    // -------- phase C: layers 8..11, store straight from registers --------
    load_cs<8, USET>(t, ct, st, ang, c, s);
    for (int r = 0; r < rmax; ++r) {
        const float* rp = lds + (unsigned)r * ROW_STRIDE;
        lds_read16<8>(rp, t, xv);
        rotate16(xv, c, s);
        float* dst = out + gbase + (long long)r * DIMX;
#pragma unroll
        for (int e = 0; e < 16; ++e)   // coalesced per e; streamed once -> NT
            __builtin_nontemporal_store(xv[e], dst + (((unsigned)e << 8) + t));
    }
}

extern "C" void kernel_launch(void* const* d_in, const int* in_sizes, int n_in,
                              void* d_out, int out_size, void* d_ws, size_t ws_size,
                              hipStream_t stream) {
    const float* x = (const float*)d_in[0];
    const float* ang = (const float*)d_in[1];
    float* out = (float*)d_out;

    int nrows = in_sizes[0] / DIMX;   // 8192 for the reference shapes

    float* ct = (float*)d_ws;
    float* st = ct + NANG;
    bool useT = (ws_size >= (size_t)(2 * NANG) * sizeof(float));

    int nwg = (nrows + ROWS_PER_WG - 1) / ROWS_PER_WG;
    size_t shmem = (size_t)ROWS_PER_WG * ROW_STRIDE * sizeof(float);  // 136 KB

    if (useT) {
        cs_table_kernel<<<(NANG + TPB - 1) / TPB, TPB, 0, stream>>>(ang, ct, st, NANG);
        butterfly_kernel<true><<<nwg, TPB, shmem, stream>>>(x, out, ct, st, ang, nrows);
    } else {
        butterfly_kernel<false><<<nwg, TPB, shmem, stream>>>(x, out, ct, st, ang, nrows);
    }
}